// HashEncoding_53721450938416
// MI455X (gfx1250) — compile-verified
//
#include <hip/hip_runtime.h>

// Instant-NGP hash-grid encoding for MI455X (gfx1250, wave32).
//
// Mapping: lane = (point, level) pair; 16 consecutive lanes share one point,
// so one wave32 = 2 points x 16 levels and its 32 float2 results form one
// contiguous 256B run -> perfectly coalesced nontemporal stores.
//
// Memory strategy: 64MB of tables are L2-resident (192MB L2); gathers use
// default RT temporal hint so they stay pinned. The 256MB output is streamed
// with TH_STORE_NT so it cannot evict the tables.

#define HE_N_LEVELS 16
#define HE_LOG2_T   19
#define HE_T_SIZE   (1u << HE_LOG2_T)
#define HE_T_MASK   (HE_T_SIZE - 1u)

#define HE_PRIME1 2654435761u
#define HE_PRIME2 805459861u

// Native clang vector (required by __builtin_nontemporal_store; HIP's float2
// is a class and is rejected).
typedef float vfloat2 __attribute__((ext_vector_type(2)));

static __device__ __forceinline__ vfloat2 lerp2(vfloat2 a, vfloat2 b, float t) {
    vfloat2 r;
    r.x = fmaf(t, b.x - a.x, a.x);
    r.y = fmaf(t, b.y - a.y, a.y);
    return r;
}

__global__ __launch_bounds__(256) void hashenc_kernel(
    const float*   __restrict__ points,   // [N,3]
    const float*   __restrict__ bb,       // [2,3]
    const vfloat2* __restrict__ tables,   // [L, T] of float2 (F=2)
    vfloat2*       __restrict__ out,      // [N, L] of float2
    int n_points)
{
    const int tid   = blockIdx.x * 256 + threadIdx.x;
    const int level = tid & (HE_N_LEVELS - 1);
    const int point = tid >> 4;
    if (point >= n_points) return;

    // res_l = floor(BASE * b^l), b = 2^(1/3)  ->  floor(2^(4 + l/3)).
    // Multiples of 3 hit exact powers of two; others are far from integer
    // boundaries, so exp2f+floorf reproduces the level-resolution table.
    const float res = floorf(exp2f(fmaf((float)level, 1.0f / 3.0f, 4.0f)));

    // 16 lanes share this point; hardware merges the identical addresses.
    const float px = points[(size_t)point * 3 + 0];
    const float py = points[(size_t)point * 3 + 1];
    const float pz = points[(size_t)point * 3 + 2];

    const float b0x = bb[0], b0y = bb[1], b0z = bb[2];
    const float b1x = bb[3], b1y = bb[4], b1z = bb[5];

    // t = (p - bb0) / ((bb1 - bb0)/res)
    const float tx = (px - b0x) * (res / (b1x - b0x));
    const float ty = (py - b0y) * (res / (b1y - b0y));
    const float tz = (pz - b0z) * (res / (b1z - b0z));

    const float fx = floorf(tx), fy = floorf(ty), fz = floorf(tz);
    const int   ix = (int)fx,    iy = (int)fy,    iz = (int)fz;
    const float wx = tx - fx,    wy = ty - fy,    wz = tz - fz;

    // Spatial hash with uint32 wraparound, matching the reference exactly.
    const unsigned hx0 = (unsigned)ix;                       // * prime0 (=1)
    const unsigned hx1 = (unsigned)(ix + 1);
    const unsigned hy0 = (unsigned)iy        * HE_PRIME1;
    const unsigned hy1 = (unsigned)(iy + 1)  * HE_PRIME1;
    const unsigned hz0 = (unsigned)iz        * HE_PRIME2;
    const unsigned hz1 = (unsigned)(iz + 1)  * HE_PRIME2;

    const vfloat2* __restrict__ tbl = tables + (size_t)level * HE_T_SIZE;

    // 8 independent 8B gathers per lane (256 in flight per wave) -> L2 hits.
    const vfloat2 e000 = tbl[(hx0 ^ hy0 ^ hz0) & HE_T_MASK];
    const vfloat2 e001 = tbl[(hx0 ^ hy0 ^ hz1) & HE_T_MASK];
    const vfloat2 e010 = tbl[(hx0 ^ hy1 ^ hz0) & HE_T_MASK];
    const vfloat2 e011 = tbl[(hx0 ^ hy1 ^ hz1) & HE_T_MASK];
    const vfloat2 e100 = tbl[(hx1 ^ hy0 ^ hz0) & HE_T_MASK];
    const vfloat2 e101 = tbl[(hx1 ^ hy0 ^ hz1) & HE_T_MASK];
    const vfloat2 e110 = tbl[(hx1 ^ hy1 ^ hz0) & HE_T_MASK];
    const vfloat2 e111 = tbl[(hx1 ^ hy1 ^ hz1) & HE_T_MASK];

    // Trilinear blend: 7 lerps x 2 features (== corner-weight product form).
    const vfloat2 c00 = lerp2(e000, e001, wz);
    const vfloat2 c01 = lerp2(e010, e011, wz);
    const vfloat2 c10 = lerp2(e100, e101, wz);
    const vfloat2 c11 = lerp2(e110, e111, wz);
    const vfloat2 c0  = lerp2(c00, c01, wy);
    const vfloat2 c1  = lerp2(c10, c11, wy);
    const vfloat2 r   = lerp2(c0,  c1,  wx);

    // Contiguous across the wave; NT so the 256MB stream doesn't evict the
    // L2-resident tables (gfx1250 temporal-hint store path).
    __builtin_nontemporal_store(r, &out[(size_t)point * HE_N_LEVELS + level]);
}

extern "C" void kernel_launch(void* const* d_in, const int* in_sizes, int n_in,
                              void* d_out, int out_size, void* d_ws, size_t ws_size,
                              hipStream_t stream) {
    const float*   points = (const float*)d_in[0];    // [N,3] f32
    const float*   bb     = (const float*)d_in[1];    // [2,3] f32
    const vfloat2* tables = (const vfloat2*)d_in[2];  // [16, 2^19, 2] f32
    vfloat2*       out    = (vfloat2*)d_out;          // [N, 16] float2 == [N,32] f32

    const int n_points = in_sizes[0] / 3;
    const long long total_threads = (long long)n_points * HE_N_LEVELS;
    const int block = 256;  // 8 waves of wave32
    const int grid  = (int)((total_threads + block - 1) / block);

    hashenc_kernel<<<grid, block, 0, stream>>>(points, bb, tables, out, n_points);
}